// PatchNCELoss_807453851903
// MI455X (gfx1250) — compile-verified
//
#include <hip/hip_runtime.h>
#include <hip/hip_bf16.h>
#include <math.h>

// PatchNCE loss for MI455X (gfx1250).
//
// S = Qraw * Kraw^T (4096 x 4096 x 256) via V_WMMA_F32_16X16X4_F32, with the
// A/B tiles DMA'd global->LDS by the Tensor Data Mover (tensor_load_to_lds,
// TENSORcnt-gated, LDS double-buffered), normalization folded into the
// epilogue as invq[m]*invk[n] scales. S (64 MB) is L2-resident (192 MB L2),
// so the per-row negative gathers in the epilogue hit L2.
// Workspace requirement: 64 MiB + 32 KiB.

typedef __attribute__((ext_vector_type(2))) float v2f;
typedef __attribute__((ext_vector_type(8))) float v8f;
typedef __attribute__((ext_vector_type(4))) unsigned int v4u;
typedef __attribute__((ext_vector_type(8))) int v8i;
typedef __attribute__((ext_vector_type(4))) int v4i;

#define N_TOTAL 4096
#define CCH     256
#define HWB     1024          // H*W per batch image
#define KNEG    128
#define INV_T   (1.0f / 0.07f)

#define BM   64
#define BN   64
#define BKK  32
#define LDTK 80               // 64-dword tile rows + 16-dword TDM pad

// ---------------------------------------------------------------- zero output
__global__ void zero_kernel(float* out) { out[0] = 0.0f; }

// ------------------------------------------------- 1/max(||x||,eps) per vector
__global__ __launch_bounds__(256) void invnorm_kernel(const float* __restrict__ feat,
                                                      float* __restrict__ inv) {
    const int n  = blockIdx.x * blockDim.x + threadIdx.x;   // 0..4095
    const int b  = n >> 10;
    const int hw = n & 1023;
    const float* p = feat + (size_t)b * (CCH * HWB) + hw;
    float s = 0.0f;
#pragma unroll 8
    for (int c = 0; c < CCH; ++c) {
        const float v = p[(size_t)c * HWB];
        s += v * v;
    }
    inv[n] = 1.0f / fmaxf(sqrtf(s), 1e-12f);
}

// ------------------------------------------------------- TDM 2D tile -> LDS
// Loads a 64(pos, contiguous) x 32(channel, stride 1024 elem) f32 tile into
// LDS, hardware-padded to 80-dword rows (pad_interval=64 dw, pad_amount=16 dw)
// so WMMA fragment reads are bank-conflict free.
__device__ __forceinline__ void tdm_load_tile(const float* gsrc, unsigned lds_off) {
    const unsigned long long ga = (unsigned long long)(size_t)gsrc;
    const v4u g0 = {
        1u,                                                  // count=1, user D#
        lds_off,                                             // lds_addr (bytes)
        (unsigned)(ga & 0xFFFFFFFFu),                        // global_addr[31:0]
        (unsigned)((ga >> 32) & 0x01FFFFFFu) | (2u << 30)    // addr[56:32] | type=2
    };
    const v8i g1 = {
        (int)((2u << 16)      // data_size = 4 bytes
            | (1u << 20)      // pad_enable
            | (5u << 22)      // pad_interval: 64 dwords
            | (15u << 25)),   // pad_amount: 16 dwords
        0,                    // atomic_barrier_addr lo | tensor_dim0 lo16 (0)
        0x4000,               // tensor_dim0 hi16 (2^30) | tensor_dim1 lo16 (0)
        0x4000 | (64 << 16),  // tensor_dim1 hi16 (2^30) | tile_dim0 = 64
        32,                   // tile_dim1 = 32, tile_dim2 = 0
        1024,                 // tensor_dim0_stride lo32 = 1024 elements
        0,                    // stride0 hi16 | stride1 lo16
        0                     // stride1 hi32
    };
    const v4i zero4 = {0, 0, 0, 0};                          // 2-D tile: groups 2/3 unused
    const v8i zero8 = {0, 0, 0, 0, 0, 0, 0, 0};
    __builtin_amdgcn_tensor_load_to_lds(g0, g1, zero4, zero4, zero8, 0);
}

// --------------------------------------------------------- WMMA f32 Gram GEMM
// 128 threads = 4 waves; 64x64 block tile; each wave owns 32x32 (2x2 v8f accs).
// Wave 0 drives the TDM double-buffer pipeline; all waves consume from LDS.
__global__ __launch_bounds__(128) void gemm_raw_wmma(const float* __restrict__ Q,
                                                     const float* __restrict__ K,
                                                     float* __restrict__ S) {
    __shared__ float As[2][BKK * LDTK];
    __shared__ float Bs[2][BKK * LDTK];

    const int tid  = threadIdx.x;
    const int wave = tid >> 5;
    const int lane = tid & 31;
    const int wm   = (wave >> 1) * 32;
    const int wn   = (wave & 1) * 32;
    const int rowbase = blockIdx.y * BM;     // 64 | 1024 => tiles never cross a batch
    const int colbase = blockIdx.x * BN;
    const int half = lane >> 4;              // 0: K pair {k,k+1}; 1: {k+2,k+3}
    const int l15  = lane & 15;

    const float* qbase = Q + (size_t)(rowbase >> 10) * (CCH * HWB) + (rowbase & 1023);
    const float* kbase = K + (size_t)(colbase >> 10) * (CCH * HWB) + (colbase & 1023);

    const unsigned asOff = (unsigned)(size_t)(&As[0][0]);    // LDS byte offsets
    const unsigned bsOff = (unsigned)(size_t)(&Bs[0][0]);
    const unsigned bufBytes = (unsigned)(BKK * LDTK * sizeof(float));

    v8f acc[2][2] = {};

    // prologue: stage k-block 0 into buffer 0
    if (wave == 0) {
        tdm_load_tile(qbase, asOff);
        tdm_load_tile(kbase, bsOff);
    }

    for (int kb = 0; kb < CCH / BKK; ++kb) {
        const int cur = kb & 1;
        if (wave == 0) {
            if (kb < CCH / BKK - 1) {
                // prefetch next k-block into the other buffer (DMA overlaps compute)
                const size_t goff = (size_t)(kb + 1) * BKK * HWB;
                tdm_load_tile(qbase + goff, asOff + (unsigned)(cur ^ 1) * bufBytes);
                tdm_load_tile(kbase + goff, bsOff + (unsigned)(cur ^ 1) * bufBytes);
                __builtin_amdgcn_s_wait_tensorcnt((short)2);  // in-order: current tile done
            } else {
                __builtin_amdgcn_s_wait_tensorcnt((short)0);
            }
        }
        __syncthreads();

        const float* A = As[cur];
        const float* B = Bs[cur];
#pragma unroll
        for (int kk = 0; kk < BKK; kk += 4) {
            const int k0 = kk + 2 * half;
            v2f a0, a1, b0, b1;
            // A 16x4 f32 layout: lane = m + 16*(k>=2), vgpr = k&1 (B mirrored with n)
            a0[0] = A[k0 * LDTK + wm + l15];
            a0[1] = A[(k0 + 1) * LDTK + wm + l15];
            a1[0] = A[k0 * LDTK + wm + 16 + l15];
            a1[1] = A[(k0 + 1) * LDTK + wm + 16 + l15];
            b0[0] = B[k0 * LDTK + wn + l15];
            b0[1] = B[(k0 + 1) * LDTK + wn + l15];
            b1[0] = B[k0 * LDTK + wn + 16 + l15];
            b1[1] = B[(k0 + 1) * LDTK + wn + 16 + l15];

            acc[0][0] = __builtin_amdgcn_wmma_f32_16x16x4_f32(false, a0, false, b0,
                          (short)0, acc[0][0], false, false);
            acc[0][1] = __builtin_amdgcn_wmma_f32_16x16x4_f32(false, a0, false, b1,
                          (short)0, acc[0][1], false, false);
            acc[1][0] = __builtin_amdgcn_wmma_f32_16x16x4_f32(false, a1, false, b0,
                          (short)0, acc[1][0], false, false);
            acc[1][1] = __builtin_amdgcn_wmma_f32_16x16x4_f32(false, a1, false, b1,
                          (short)0, acc[1][1], false, false);
        }
        __syncthreads();   // all waves done reading buf[cur] before it is refilled
    }

    // D 16x16 f32 layout: VGPR v -> M = v + 8*half, N = lane&15
#pragma unroll
    for (int i = 0; i < 2; ++i) {
#pragma unroll
        for (int j = 0; j < 2; ++j) {
            const int m0 = rowbase + wm + 16 * i + 8 * half;
            const int n0 = colbase + wn + 16 * j + l15;
#pragma unroll
            for (int v = 0; v < 8; ++v) {
                S[(size_t)(m0 + v) * N_TOTAL + n0] = acc[i][j][v];
            }
        }
    }
}

// --------------------------------------------------------------- loss epilogue
// One wave32 per row: gather 128 negatives from the L2-resident Gram matrix,
// apply folded normalization, exp-sum, shuffle-reduce, atomicAdd. Logits are
// bounded by 1/0.07 so no max-shift is needed for f32 logsumexp.
__global__ __launch_bounds__(256) void nce_loss_kernel(const float* __restrict__ S,
                                                       const float* __restrict__ invq,
                                                       const float* __restrict__ invk,
                                                       const int* __restrict__ negidx,
                                                       float* __restrict__ out) {
    const int wave = threadIdx.x >> 5;
    const int lane = threadIdx.x & 31;
    const int n = blockIdx.x * 8 + wave;

    const float* Srow = S + (size_t)n * N_TOTAL;
    const float qscale = invq[n] * INV_T;
    const float lpos = Srow[n] * invk[n] * qscale;

    float local = 0.0f;
#pragma unroll
    for (int jj = 0; jj < 4; ++jj) {
        const int j = lane + 32 * jj;
        int id = negidx[n * KNEG + j];
        id += (id >= n) ? 1 : 0;          // self-exclusion shift
        local += expf(Srow[id] * invk[id] * qscale);
    }
#pragma unroll
    for (int off = 16; off > 0; off >>= 1)
        local += __shfl_xor(local, off, 32);

    if (lane == 0) {
        const float total = local + expf(lpos);
        const float row_loss = logf(total) - lpos;
        atomicAdd(out, row_loss * (1.0f / N_TOTAL));
    }
}

// ------------------------------------------------------------------- launcher
extern "C" void kernel_launch(void* const* d_in, const int* in_sizes, int n_in,
                              void* d_out, int out_size, void* d_ws, size_t ws_size,
                              hipStream_t stream) {
    const float* feat_q = (const float*)d_in[0];
    const float* feat_k = (const float*)d_in[1];
    const int*   negidx = (const int*)d_in[2];
    float* out = (float*)d_out;

    char* ws = (char*)d_ws;
    float* S    = (float*)ws;                                          // 64 MiB
    float* invq = (float*)(ws + (size_t)N_TOTAL * N_TOTAL * sizeof(float));
    float* invk = invq + N_TOTAL;

    zero_kernel<<<1, 1, 0, stream>>>(out);
    invnorm_kernel<<<N_TOTAL / 256, 256, 0, stream>>>(feat_q, invq);
    invnorm_kernel<<<N_TOTAL / 256, 256, 0, stream>>>(feat_k, invk);

    dim3 grid(N_TOTAL / BN, N_TOTAL / BM);
    gemm_raw_wmma<<<grid, 128, 0, stream>>>(feat_q, feat_k, S);

    nce_loss_kernel<<<N_TOTAL / 8, 256, 0, stream>>>(S, invq, invk, negidx, out);
}